// Post_rois_42193758716031
// MI455X (gfx1250) — compile-verified
//
#include <hip/hip_runtime.h>
#include <math.h>

#define B_IMGS   32
#define N_PRI    32768
#define NCLS     21
#define TOPK     200
#define CONF_TH  0.01f
#define IOU_TH   0.4f
#define VAR0     0.1f
#define VAR1     0.2f

#define TPB      1024                 // 32 waves (wave32) per image
#define NGRP     8                    // element i = g*4096 + tid*4 + j
#define EPT      32                   // 8 groups x 4 contiguous scores per thread
#define NWAVES   (TPB / 32)

// gfx1250 async VMEM->LDS path (ASYNCcnt-tracked), if this toolchain exposes it.
#if defined(__has_builtin)
#if __has_builtin(__builtin_amdgcn_global_load_async_to_lds_b128) && \
    __has_builtin(__builtin_amdgcn_s_wait_asynccnt)
#define USE_ASYNC_LDS 1
#endif
#endif

#ifdef USE_ASYNC_LDS
typedef int v4i __attribute__((ext_vector_type(4)));
typedef __attribute__((address_space(1))) v4i* async_gptr_t;   // global (AS1)
typedef __attribute__((address_space(3))) v4i* async_lptr_t;   // LDS (AS3)
#endif

// ---------------------------------------------------------------------------
// Kernel A: fused softmax(class 1) + box decode + conf-threshold.
// Pure streaming: ~125 MB @ 23.3 TB/s -> ~5us; memory-bound by design.
// ---------------------------------------------------------------------------
__global__ __launch_bounds__(256) void decode_score_kernel(
    const float* __restrict__ loc,      // (B*N, 4)
    const float* __restrict__ conf,     // (B*N, 21)
    const float* __restrict__ prior,    // (N, 4)
    float*  __restrict__ scores,        // (B*N)
    float4* __restrict__ boxes)         // (B*N)
{
    const int r = blockIdx.x * blockDim.x + threadIdx.x;
    if (r >= B_IMGS * N_PRI) return;

    const float* c = conf + (size_t)r * NCLS;
    // speculative prefetch ahead -> global_prefetch_b8
    __builtin_prefetch(c + 64 * NCLS, 0, 1);

    // numerically-stable softmax, keep only class 1
    float m = c[0];
#pragma unroll
    for (int i = 1; i < NCLS; ++i) m = fmaxf(m, c[i]);
    float sum = 0.0f;
    float e1  = 0.0f;
#pragma unroll
    for (int i = 0; i < NCLS; ++i) {
        float e = expf(c[i] - m);
        sum += e;
        if (i == 1) e1 = e;
    }
    const float s1 = e1 / sum;
    scores[r] = (s1 > CONF_TH) ? s1 : -__builtin_inff();

    // box decode
    const int n = r & (N_PRI - 1);
    const float4 l = ((const float4*)loc)[r];
    const float4 p = ((const float4*)prior)[n];
    const float cx = p.x + l.x * VAR0 * p.z;
    const float cy = p.y + l.y * VAR0 * p.w;
    const float w  = p.z * expf(l.z * VAR1);
    const float h  = p.w * expf(l.w * VAR1);
    float4 b;
    b.x = fminf(fmaxf(cx - 0.5f * w, 0.0f), 1.0f);
    b.y = fminf(fmaxf(cy - 0.5f * h, 0.0f), 1.0f);
    b.z = fminf(fmaxf(cx + 0.5f * w, 0.0f), 1.0f);
    b.w = fminf(fmaxf(cy + 0.5f * h, 0.0f), 1.0f);
    boxes[r] = b;
}

// ---------------------------------------------------------------------------
// Kernel B: per-image sequential NMS. One 1024-thread block per image.
// Element ownership: i = g*4096 + tid*4 + j (g<8, j<4) -> each lane owns
// 8 groups of 4 *contiguous* elements. Scores live in VGPRs (32/lane, well
// under the 8-wave/SIMD VGPR cap -> no scratch spills); boxes are re-read
// per iteration as coalesced b128 loads from L2-resident workspace, with
// already-suppressed lanes exec-masked off the load. Prologue stages scores
// through LDS via the gfx1250 async DMA path (8x b128, ASYNCcnt-tracked).
// Argmax: unrolled register scan -> wave32 shuffle butterfly -> 32-entry
// LDS reduce. 200 serial iterations, 2 barriers each.
// ---------------------------------------------------------------------------
__global__ __launch_bounds__(TPB) void nms_kernel(
    const float*  __restrict__ scores_g,
    const float4* __restrict__ boxes_g,
    float*        __restrict__ out)      // (B, 1, TOPK, 5)
{
    const int img  = blockIdx.x;
    const int tid  = threadIdx.x;
    const int lane = tid & 31;
    const int wave = tid >> 5;

    __shared__ float  red_s[NWAVES];
    __shared__ int    red_i[NWAVES];
    __shared__ float  sel_score;
    __shared__ int    sel_idx;

    const float NEG_INF = -__builtin_inff();

    float s[EPT];                      // s[g*4+j] <-> element g*4096+tid*4+j
    const size_t base = (size_t)img * N_PRI;

#ifdef USE_ASYNC_LDS
    // Stage each lane's 4 contiguous scores per group through LDS with the
    // async DMA path (global_load_async_to_lds_b128), then ds_load into
    // VGPRs. Constant indices -> guaranteed SROA promotion. Barriers guard
    // staging-buffer reuse (DMA write vs ds_load WAR across queues).
    __shared__ float4 stage[TPB];
#pragma unroll
    for (int g = 0; g < NGRP; ++g) {
        __builtin_amdgcn_global_load_async_to_lds_b128(
            (async_gptr_t)(scores_g + base + g * 4096 + tid * 4),
            (async_lptr_t)&stage[tid],
            0, 0);
        __builtin_amdgcn_s_wait_asynccnt(0);
        __syncthreads();
        const float4 sv = stage[tid];
        s[g * 4 + 0] = sv.x;
        s[g * 4 + 1] = sv.y;
        s[g * 4 + 2] = sv.z;
        s[g * 4 + 3] = sv.w;
        __syncthreads();
    }
#else
#pragma unroll
    for (int g = 0; g < NGRP; ++g) {
        const float4 sv = ((const float4*)(scores_g + base))[g * 1024 + tid];
        s[g * 4 + 0] = sv.x;
        s[g * 4 + 1] = sv.y;
        s[g * 4 + 2] = sv.z;
        s[g * 4 + 3] = sv.w;
    }
#endif

    for (int it = 0; it < TOPK; ++it) {
        // --- per-thread argmax over register chunk (first-max = lowest idx) ---
        float bs = s[0];
        int   bi = tid * 4;            // i(g=0,j=0)
#pragma unroll
        for (int k = 1; k < EPT; ++k) {
            const int i = (k >> 2) * 4096 + tid * 4 + (k & 3);
            if (s[k] > bs) { bs = s[k]; bi = i; }
        }
        // --- wave32 butterfly reduce (max score, tie -> lower index) ---
#pragma unroll
        for (int off = 16; off > 0; off >>= 1) {
            const float os = __shfl_xor(bs, off, 32);
            const int   oi = __shfl_xor(bi, off, 32);
            if (os > bs || (os == bs && oi < bi)) { bs = os; bi = oi; }
        }
        if (lane == 0) { red_s[wave] = bs; red_i[wave] = bi; }
        __syncthreads();                              // (1) red_* visible
        if (tid == 0) {
            float gs = red_s[0];
            int   gi = red_i[0];
#pragma unroll
            for (int w = 1; w < NWAVES; ++w) {
                if (red_s[w] > gs || (red_s[w] == gs && red_i[w] < gi)) {
                    gs = red_s[w]; gi = red_i[w];
                }
            }
            sel_score = gs;
            sel_idx   = gi;
        }
        __syncthreads();                              // (2) sel_* visible

        const int   gi     = sel_idx;
        const float gscore = sel_score;
        const bool  valid  = (gscore != NEG_INF);

        // owner kills its own score (matches s.at[idx].set(-inf))
#pragma unroll
        for (int k = 0; k < EPT; ++k) {
            if (gi == (k >> 2) * 4096 + tid * 4 + (k & 3)) s[k] = NEG_INF;
        }

        if (tid == 0) {
            float* row = out + ((size_t)img * TOPK + it) * 5;
            if (valid) {
                const float4 Bb0 = boxes_g[base + gi];
                row[0] = (float)img;
                row[1] = Bb0.x; row[2] = Bb0.y; row[3] = Bb0.z; row[4] = Bb0.w;
            } else {
                row[0] = 0.0f; row[1] = 0.0f; row[2] = 0.0f; row[3] = 0.0f; row[4] = 0.0f;
            }
        }

        if (valid) {
            // uniform broadcast load of the selected box (single L2 line)
            const float4 Bb = boxes_g[base + gi];
            const float a1 = (Bb.z - Bb.x) * (Bb.w - Bb.y);
#pragma unroll
            for (int k = 0; k < EPT; ++k) {
                if (s[k] != NEG_INF) {      // suppressed lanes skip the load
                    const float4 b2 =
                        boxes_g[base + (k >> 2) * 4096 + tid * 4 + (k & 3)];
                    const float ltx = fmaxf(Bb.x, b2.x);
                    const float lty = fmaxf(Bb.y, b2.y);
                    const float rbx = fminf(Bb.z, b2.z);
                    const float rby = fminf(Bb.w, b2.w);
                    const float iw  = fmaxf(rbx - ltx, 0.0f);
                    const float ih  = fmaxf(rby - lty, 0.0f);
                    const float inter = iw * ih;
                    const float a2  = (b2.z - b2.x) * (b2.w - b2.y);
                    const float den = a1 + a2 - inter;
                    const float iou = (den > 0.0f) ? (inter / den) : 0.0f;
                    if (iou > IOU_TH) s[k] = NEG_INF;
                }
            }
        }
        // next iteration's red_* writes happen after (2); thread 0's read of
        // red_* finished before (2); sel_* rewritten only after next (1).
    }
}

// ---------------------------------------------------------------------------
// Host launcher. Inputs (setup_inputs order): img, loc_data, conf_data,
// prior_data. Output: (B,1,TOPK,5) float32 = 32000 floats.
// Workspace: [0, 4MB) scores, [4MB, 20MB) boxes(float4).
// ---------------------------------------------------------------------------
extern "C" void kernel_launch(void* const* d_in, const int* in_sizes, int n_in,
                              void* d_out, int out_size, void* d_ws, size_t ws_size,
                              hipStream_t stream) {
    (void)in_sizes; (void)n_in; (void)out_size; (void)ws_size;
    const float* loc   = (const float*)d_in[1];
    const float* conf  = (const float*)d_in[2];
    const float* prior = (const float*)d_in[3];
    float* outp = (float*)d_out;

    float*  scores = (float*)d_ws;
    float4* boxes  = (float4*)((char*)d_ws + (size_t)B_IMGS * N_PRI * sizeof(float));

    const int rows = B_IMGS * N_PRI;
    decode_score_kernel<<<(rows + 255) / 256, 256, 0, stream>>>(
        loc, conf, prior, scores, boxes);
    nms_kernel<<<B_IMGS, TPB, 0, stream>>>(scores, boxes, outp);
}